// CSimLoss_41180146434815
// MI455X (gfx1250) — compile-verified
//
#include <hip/hip_runtime.h>
#include <hip/hip_bf16.h>

typedef __attribute__((ext_vector_type(2))) float v2f;
typedef __attribute__((ext_vector_type(8))) float v8f;

#define TLEN 512
#define XPAD_FRONT 16
#define XBUF 768            // 16 front pad + 512 data + 240 back pad (max idx used: 751)
#define PBUF 513
#define ROW_STRIDE 1296     // floats per row of LDS (768 + 513 = 1281, padded)
#define WAVES_PER_BLOCK 8

__global__ void CSim_zero_out(float* out) { out[0] = 0.0f; }

__global__ __launch_bounds__(256) void CSim_wmma_kernel(const float* __restrict__ x,
                                                        float* __restrict__ out,
                                                        float inv_scale) {
    __shared__ float smem[WAVES_PER_BLOCK * ROW_STRIDE];

    const int lane = threadIdx.x & 31;
    const int wv   = threadIdx.x >> 5;
    const int row  = blockIdx.x * WAVES_PER_BLOCK + wv;

    float* xbuf = &smem[wv * ROW_STRIDE];  // 768 floats, padded row buffer
    float* xb   = xbuf + XPAD_FRONT;       // logical x[0]; xb[-15..-1] is zero pad
    float* Pb   = xbuf + XBUF;             // 513 floats: P[0..512] prefix sums of x^2

    const int l15 = lane & 15;
    const int hi  = lane >> 4;             // 0 or 1

    // ---- load row into LDS (coalesced b128), zero the pads ----
    const float* xrow = x + (size_t)row * TLEN;
    const float4* xrow4 = (const float4*)xrow;
    float4* xb4 = (float4*)xb;
    #pragma unroll
    for (int j = 0; j < 4; ++j)
        xb4[j * 32 + lane] = xrow4[j * 32 + lane];
    if (lane < XPAD_FRONT) xbuf[lane] = 0.0f;
    #pragma unroll
    for (int j = lane; j < XBUF - XPAD_FRONT - TLEN; j += 32)
        xb[TLEN + j] = 0.0f;
    __syncthreads();

    // ---- prefix sums of squares: P[j] = sum_{t<j} x[t]^2 ----
    float p[16];
    float run = 0.0f;
    #pragma unroll
    for (int j = 0; j < 16; ++j) {
        float v = xb[lane * 16 + j];
        run += v * v;
        p[j] = run;
    }
    float incl = run;
    #pragma unroll
    for (int d = 1; d < 32; d <<= 1) {
        float t = __shfl_up(incl, d, 32);
        if (lane >= d) incl += t;
    }
    float excl = incl - run;
    if (lane == 0) Pb[0] = 0.0f;
    #pragma unroll
    for (int j = 0; j < 16; ++j) Pb[lane * 16 + 1 + j] = excl + p[j];
    __syncthreads();
    const float total = Pb[TLEN];

    // ---- WMMA: D2[r,m] = sum_t x[t-r] * x[t+16m] == lagged dot for lag 16m+r ----
    // A fragment (Toeplitz): per-lane contiguous stream starting at 2*hi - l15
    // B fragment:            per-lane contiguous stream starting at 2*hi + 16*(l15+m0)
    v8f acc1 = {};   // m0 = 0  -> lags 0..255
    v8f acc2 = {};   // m0 = 16 -> lags 256..511
    const int a_base  = 2 * hi - l15;
    const int b_base  = 2 * hi + 16 * l15;

    #pragma unroll 4
    for (int k = 0; k < 64; ++k) {
        const int ai = a_base + 4 * k;
        const int bi = b_base + 4 * k;
        v2f A;  A.x  = xb[ai];        A.y  = xb[ai + 1];
        v2f B1; B1.x = xb[bi];        B1.y = xb[bi + 1];
        v2f B2; B2.x = xb[bi + 256];  B2.y = xb[bi + 257];
        acc1 = __builtin_amdgcn_wmma_f32_16x16x4_f32(false, A, false, B1,
                                                     (short)0, acc1, false, false);
        acc2 = __builtin_amdgcn_wmma_f32_16x16x4_f32(false, A, false, B2,
                                                     (short)0, acc2, false, false);
    }
    #pragma unroll 4
    for (int k = 64; k < 128; ++k) {
        const int ai = a_base + 4 * k;
        const int bi = b_base + 4 * k;
        v2f A;  A.x  = xb[ai]; A.y  = xb[ai + 1];
        v2f B1; B1.x = xb[bi]; B1.y = xb[bi + 1];
        acc1 = __builtin_amdgcn_wmma_f32_16x16x4_f32(false, A, false, B1,
                                                     (short)0, acc1, false, false);
    }

    // ---- normalize and accumulate: element (VGPR g, lane) holds lag 16*(m0+l15)+g+8*hi ----
    float s = 0.0f;
    const float eps = 1e-8f;
    #pragma unroll
    for (int g = 0; g < 8; ++g) {
        const int i1 = 16 * l15 + g + 8 * hi;          // 0..255 (skip lag 0)
        if (i1 >= 1) {
            float den = fmaxf(sqrtf(Pb[TLEN - i1]) * sqrtf(total - Pb[i1]), eps);
            s += acc1[g] / den;
        }
        const int i2 = 256 + 16 * l15 + g + 8 * hi;    // 256..511 (skip lag 511)
        if (i2 <= TLEN - 2) {
            float den = fmaxf(sqrtf(Pb[TLEN - i2]) * sqrtf(total - Pb[i2]), eps);
            s += acc2[g] / den;
        }
    }
    #pragma unroll
    for (int d = 16; d >= 1; d >>= 1) s += __shfl_xor(s, d, 32);
    if (lane == 0) atomicAdd(out, s * inv_scale);
}

extern "C" void kernel_launch(void* const* d_in, const int* in_sizes, int n_in,
                              void* d_out, int out_size, void* d_ws, size_t ws_size,
                              hipStream_t stream) {
    const float* x = (const float*)d_in[0];
    float* out = (float*)d_out;

    const int rows = in_sizes[0] / TLEN;               // B*C = 16384
    const float inv_scale = 1.0f / ((float)rows * (float)(TLEN - 2));

    hipLaunchKernelGGL(CSim_zero_out, dim3(1), dim3(1), 0, stream, out);
    hipLaunchKernelGGL(CSim_wmma_kernel,
                       dim3(rows / WAVES_PER_BLOCK), dim3(32 * WAVES_PER_BLOCK),
                       0, stream, x, out, inv_scale);
}